// ICPRegistration_1924145348952
// MI455X (gfx1250) — compile-verified
//
#include <hip/hip_runtime.h>
#include <math.h>

typedef float v2f __attribute__((ext_vector_type(2)));
typedef float v8f __attribute__((ext_vector_type(8)));

constexpr int BATCH = 4;
constexpr int NPTS  = 4096;
constexpr int MPTS  = 4096;
constexpr int ITERS = 10;
constexpr int MTILES = MPTS / 16;   // 256

// Workspace layout (floats)
constexpr int WS_SRC = 0;                     // transformed source [B,N,3]
constexpr int WS_ACC = BATCH * NPTS * 3;      // 15 accumulators per batch
constexpr int WS_R   = WS_ACC + BATCH * 15;   // R [B,3,3]
constexpr int WS_T   = WS_R + BATCH * 9;      // t [B,3]
constexpr int WS_TA  = WS_T + BATCH * 3;      // packed tgt A-operands [B,256,32,2]

// Output layout (floats): R [B,3,3] @0, t [B,3] @36, transformed [B,N,3] @48
constexpr int OUT_T   = BATCH * 9;
constexpr int OUT_PTS = BATCH * 9 + BATCH * 3;

__device__ __forceinline__ unsigned umin32(unsigned a, unsigned b) { return a < b ? a : b; }

__global__ void init_src_kernel(const float* __restrict__ src, float* __restrict__ ws) {
    int i = blockIdx.x * blockDim.x + threadIdx.x;
    if (i < BATCH * NPTS * 3) ws[WS_SRC + i] = src[i];
}

// Precompute per-lane WMMA A-operand for every target tile (tgt is static
// across all ICP iterations). Lane l of tile handles A row m = tile*16+(l&15):
//   lanes 0-15  -> (K=0,K=1) = (-2tx, -2ty)
//   lanes 16-31 -> (K=2,K=3) = (-2tz, |t|^2)
__global__ void pack_tgt_kernel(const float* __restrict__ tgt, float* __restrict__ ws) {
    const int i = blockIdx.x * blockDim.x + threadIdx.x;   // B*256*32 slots
    if (i >= BATCH * MTILES * 32) return;
    const int lane = i & 31;
    const int tile = (i >> 5) & (MTILES - 1);
    const int b    = i >> 13;
    const int m    = tile * 16 + (lane & 15);
    const float tx = tgt[((size_t)b * MPTS + m) * 3 + 0];
    const float ty = tgt[((size_t)b * MPTS + m) * 3 + 1];
    const float tz = tgt[((size_t)b * MPTS + m) * 3 + 2];
    const float tq = tx * tx + ty * ty + tz * tz;
    float a0 = (lane < 16) ? (-2.0f * tx) : (-2.0f * tz);
    float a1 = (lane < 16) ? (-2.0f * ty) : tq;
    ws[WS_TA + (size_t)i * 2 + 0] = a0;
    ws[WS_TA + (size_t)i * 2 + 1] = a1;
}

__global__ void zero_acc_kernel(float* __restrict__ ws) {
    int i = threadIdx.x;
    if (i < BATCH * 15) ws[WS_ACC + i] = 0.0f;
}

// One wave = one 16-src-point tile; sweeps all M/16 target tiles.
// WMMA computes the full squared distance (|s|^2 comes in via the C operand);
// argmin is a u32 min-tree over (distance | index) packed keys. The key mask
// clears the sign bit: a tiny-negative rounded distance (true d ~ 0) becomes a
// tiny positive key, which ranks best -- exactly the desired semantics and
// zero clamp instructions.
__global__ __launch_bounds__(256)
void nn_accum_kernel(const float* __restrict__ tgt, float* __restrict__ ws) {
    __shared__ float acc[15];

    const int wave = threadIdx.x >> 5;
    const int lane = threadIdx.x & 31;
    const int b    = blockIdx.x >> 5;                      // 32 blocks per batch
    const int srcTile = (blockIdx.x & 31) * 8 + wave;      // 256 tiles per batch
    const int n    = lane & 15;                            // src column within tile
    const int srcIdx = srcTile * 16 + n;

    const float* __restrict__ srcB = ws + WS_SRC + (size_t)b * NPTS * 3;
    const float* __restrict__ tgtB = tgt + (size_t)b * MPTS * 3;
    const v2f*   __restrict__ taB  = (const v2f*)(ws + WS_TA) + (size_t)b * MTILES * 32;

    // B operand (4x16, K x N): column n = [sx, sy, sz, 1]^T
    const float sx = srcB[srcIdx * 3 + 0];
    const float sy = srcB[srcIdx * 3 + 1];
    const float sz = srcB[srcIdx * 3 + 2];
    v2f bm;
    bm[0] = (lane < 16) ? sx : sz;      // K=0 (low lanes) / K=2 (high lanes)
    bm[1] = (lane < 16) ? sy : 1.0f;    // K=1 (low lanes) / K=3 (high lanes)

    // C operand: |s|^2 for this lane's column in every row slot -> WMMA emits
    // the full squared distance  |s|^2 - 2 s.t + |t|^2  >= 0 (up to rounding).
    const float q = sx * sx + sy * sy + sz * sz;
    const v8f c0 = { q, q, q, q, q, q, q, q };

    const int rowOff = (lane >= 16) ? 8 : 0;
    unsigned best = 0xFFFFFFFFu;

    for (int tile = 0; tile < MTILES; ++tile) {
        const v2f am = taB[tile * 32 + lane];

        v8f c = __builtin_amdgcn_wmma_f32_16x16x4_f32(
                /*neg_a=*/false, am, /*neg_b=*/false, bm,
                /*c_mod=*/(short)0, c0, /*reuse_a=*/false, /*reuse_b=*/false);

        const unsigned base = (unsigned)(tile * 16 + rowOff);
        // pack (|distance| | row index); sign bit cleared by the mask
        unsigned k0 = (__float_as_uint(c[0]) & 0x7FFFF000u) | (base + 0);
        unsigned k1 = (__float_as_uint(c[1]) & 0x7FFFF000u) | (base + 1);
        unsigned k2 = (__float_as_uint(c[2]) & 0x7FFFF000u) | (base + 2);
        unsigned k3 = (__float_as_uint(c[3]) & 0x7FFFF000u) | (base + 3);
        unsigned k4 = (__float_as_uint(c[4]) & 0x7FFFF000u) | (base + 4);
        unsigned k5 = (__float_as_uint(c[5]) & 0x7FFFF000u) | (base + 5);
        unsigned k6 = (__float_as_uint(c[6]) & 0x7FFFF000u) | (base + 6);
        unsigned k7 = (__float_as_uint(c[7]) & 0x7FFFF000u) | (base + 7);
        // min tree (short dependency chain, min3-friendly)
        const unsigned m01 = umin32(k0, k1);
        const unsigned m23 = umin32(k2, k3);
        const unsigned m45 = umin32(k4, k5);
        const unsigned m67 = umin32(k6, k7);
        const unsigned m03 = umin32(m01, m23);
        const unsigned m47 = umin32(m45, m67);
        best = umin32(best, umin32(m03, m47));
    }

    // Combine the two column halves (rows 0-7 vs 8-15) across lane pairs.
    best = umin32(best, (unsigned)__shfl_xor((int)best, 16, 32));
    const int mini = (int)(best & 0xFFFu);

    if (threadIdx.x < 15) acc[threadIdx.x] = 0.0f;
    __syncthreads();

    if (lane < 16) {
        const float mx = tgtB[mini * 3 + 0];
        const float my = tgtB[mini * 3 + 1];
        const float mz = tgtB[mini * 3 + 2];
        atomicAdd(&acc[0], sx);
        atomicAdd(&acc[1], sy);
        atomicAdd(&acc[2], sz);
        atomicAdd(&acc[3], mx);
        atomicAdd(&acc[4], my);
        atomicAdd(&acc[5], mz);
        atomicAdd(&acc[6],  sx * mx);
        atomicAdd(&acc[7],  sx * my);
        atomicAdd(&acc[8],  sx * mz);
        atomicAdd(&acc[9],  sy * mx);
        atomicAdd(&acc[10], sy * my);
        atomicAdd(&acc[11], sy * mz);
        atomicAdd(&acc[12], sz * mx);
        atomicAdd(&acc[13], sz * my);
        atomicAdd(&acc[14], sz * mz);
    }
    __syncthreads();

    if (threadIdx.x < 15) atomicAdd(&ws[WS_ACC + b * 15 + threadIdx.x], acc[threadIdx.x]);
}

// 3x3 Kabsch: Jacobi eigendecomposition of H^T H -> V, U = H V S^-1,
// R = V U^T with det sign fix on V's 3rd column (matches svd desc order).
__global__ void kabsch_kernel(float* __restrict__ ws, float* __restrict__ out, int writeOut) {
    const int b = threadIdx.x;
    if (b >= BATCH) return;

    const float invN = 1.0f / (float)NPTS;
    const float* a = ws + WS_ACC + b * 15;
    float cs[3] = { a[0] * invN, a[1] * invN, a[2] * invN };
    float ct[3] = { a[3] * invN, a[4] * invN, a[5] * invN };
    float Hm[3][3];
    for (int d = 0; d < 3; ++d)
        for (int e = 0; e < 3; ++e)
            Hm[d][e] = a[6 + d * 3 + e] * invN - cs[d] * ct[e];

    // S = H^T H (symmetric)
    float A[3][3];
    for (int i = 0; i < 3; ++i)
        for (int j = 0; j < 3; ++j) {
            float s = 0.0f;
            for (int k = 0; k < 3; ++k) s += Hm[k][i] * Hm[k][j];
            A[i][j] = s;
        }

    float V[3][3] = { {1,0,0}, {0,1,0}, {0,0,1} };
    for (int sweep = 0; sweep < 16; ++sweep) {
        for (int pi = 0; pi < 3; ++pi) {
            const int p = (pi == 2) ? 1 : 0;
            const int q = (pi == 0) ? 1 : 2;
            const float apq = A[p][q];
            if (fabsf(apq) < 1e-30f) continue;
            const float theta = (A[q][q] - A[p][p]) / (2.0f * apq);
            const float tj = ((theta >= 0.0f) ? 1.0f : -1.0f) /
                             (fabsf(theta) + sqrtf(theta * theta + 1.0f));
            const float cj = 1.0f / sqrtf(tj * tj + 1.0f);
            const float sj = tj * cj;
            for (int k = 0; k < 3; ++k) {
                const float akp = A[k][p], akq = A[k][q];
                A[k][p] = cj * akp - sj * akq;
                A[k][q] = sj * akp + cj * akq;
            }
            for (int k = 0; k < 3; ++k) {
                const float apk = A[p][k], aqk = A[q][k];
                A[p][k] = cj * apk - sj * aqk;
                A[q][k] = sj * apk + cj * aqk;
            }
            for (int k = 0; k < 3; ++k) {
                const float vkp = V[k][p], vkq = V[k][q];
                V[k][p] = cj * vkp - sj * vkq;
                V[k][q] = sj * vkp + cj * vkq;
            }
        }
    }

    float w[3] = { A[0][0], A[1][1], A[2][2] };
    for (int i = 0; i < 2; ++i)
        for (int j = 0; j < 2 - i; ++j)
            if (w[j] < w[j + 1]) {
                const float tw = w[j]; w[j] = w[j + 1]; w[j + 1] = tw;
                for (int k = 0; k < 3; ++k) {
                    const float tv = V[k][j]; V[k][j] = V[k][j + 1]; V[k][j + 1] = tv;
                }
            }

    float sv[3];
    for (int i = 0; i < 3; ++i) sv[i] = sqrtf(fmaxf(w[i], 0.0f));

    float U[3][3];
    for (int i = 0; i < 2; ++i) {
        const float inv = 1.0f / fmaxf(sv[i], 1e-20f);
        for (int k = 0; k < 3; ++k)
            U[k][i] = (Hm[k][0] * V[0][i] + Hm[k][1] * V[1][i] + Hm[k][2] * V[2][i]) * inv;
    }
    if (sv[2] > 1e-6f * fmaxf(sv[0], 1e-20f)) {
        const float inv = 1.0f / sv[2];
        for (int k = 0; k < 3; ++k)
            U[k][2] = (Hm[k][0] * V[0][2] + Hm[k][1] * V[1][2] + Hm[k][2] * V[2][2]) * inv;
    } else {
        U[0][2] = U[1][0] * U[2][1] - U[2][0] * U[1][1];
        U[1][2] = U[2][0] * U[0][1] - U[0][0] * U[2][1];
        U[2][2] = U[0][0] * U[1][1] - U[1][0] * U[0][1];
    }

    float R[3][3];
    for (int i = 0; i < 3; ++i)
        for (int j = 0; j < 3; ++j)
            R[i][j] = V[i][0] * U[j][0] + V[i][1] * U[j][1] + V[i][2] * U[j][2];

    const float det = R[0][0] * (R[1][1] * R[2][2] - R[1][2] * R[2][1])
                    - R[0][1] * (R[1][0] * R[2][2] - R[1][2] * R[2][0])
                    + R[0][2] * (R[1][0] * R[2][1] - R[1][1] * R[2][0]);
    if (det < 0.0f) {
        for (int k = 0; k < 3; ++k) V[k][2] = -V[k][2];
        for (int i = 0; i < 3; ++i)
            for (int j = 0; j < 3; ++j)
                R[i][j] = V[i][0] * U[j][0] + V[i][1] * U[j][1] + V[i][2] * U[j][2];
    }

    float t[3];
    for (int i = 0; i < 3; ++i)
        t[i] = ct[i] - (R[i][0] * cs[0] + R[i][1] * cs[1] + R[i][2] * cs[2]);

    for (int i = 0; i < 3; ++i)
        for (int j = 0; j < 3; ++j)
            ws[WS_R + b * 9 + i * 3 + j] = R[i][j];
    for (int i = 0; i < 3; ++i) ws[WS_T + b * 3 + i] = t[i];

    if (writeOut) {
        for (int i = 0; i < 3; ++i)
            for (int j = 0; j < 3; ++j)
                out[b * 9 + i * 3 + j] = R[i][j];
        for (int i = 0; i < 3; ++i) out[OUT_T + b * 3 + i] = t[i];
    }
}

// x' = x @ R + t (R applied on the right, faithful to the reference)
__global__ void transform_kernel(float* __restrict__ ws, float* __restrict__ out, int writeOut) {
    const int i = blockIdx.x * blockDim.x + threadIdx.x;
    if (i >= BATCH * NPTS) return;
    const int b = i / NPTS;
    float* p = ws + WS_SRC + (size_t)i * 3;
    const float* R = ws + WS_R + b * 9;
    const float* t = ws + WS_T + b * 3;
    const float x = p[0], y = p[1], z = p[2];
    const float nx = x * R[0] + y * R[3] + z * R[6] + t[0];
    const float ny = x * R[1] + y * R[4] + z * R[7] + t[1];
    const float nz = x * R[2] + y * R[5] + z * R[8] + t[2];
    p[0] = nx; p[1] = ny; p[2] = nz;
    if (writeOut) {
        out[OUT_PTS + (size_t)i * 3 + 0] = nx;
        out[OUT_PTS + (size_t)i * 3 + 1] = ny;
        out[OUT_PTS + (size_t)i * 3 + 2] = nz;
    }
}

extern "C" void kernel_launch(void* const* d_in, const int* in_sizes, int n_in,
                              void* d_out, int out_size, void* d_ws, size_t ws_size,
                              hipStream_t stream) {
    const float* src = (const float*)d_in[0];
    const float* tgt = (const float*)d_in[1];
    float* out = (float*)d_out;
    float* ws  = (float*)d_ws;

    init_src_kernel<<<(BATCH * NPTS * 3 + 255) / 256, 256, 0, stream>>>(src, ws);
    pack_tgt_kernel<<<(BATCH * MTILES * 32 + 255) / 256, 256, 0, stream>>>(tgt, ws);

    for (int it = 0; it < ITERS; ++it) {
        const int fin = (it == ITERS - 1) ? 1 : 0;
        zero_acc_kernel<<<1, 64, 0, stream>>>(ws);
        nn_accum_kernel<<<BATCH * (NPTS / 128), 256, 0, stream>>>(tgt, ws);
        kabsch_kernel<<<1, 32, 0, stream>>>(ws, out, fin);
        transform_kernel<<<(BATCH * NPTS + 255) / 256, 256, 0, stream>>>(ws, out, fin);
    }
}